// rate_loss_3985729650886
// MI455X (gfx1250) — compile-verified
//
#include <hip/hip_runtime.h>
#include <hip/hip_bf16.h>

// ---------------- problem constants (match reference) ----------------
#define BPU     5
#define NBINS   200            // NBINS_MAX
#define HLEN    196            // NBINS - BPU + 1  (hist_add length)
#define GLEN    195            // NBINS - BPU      (g length)

// intermediate offset-histogram of floor(x*5): index = floor(x*5)+IOFF
#define NI      2048
#define IOFF    1024

#define BLOCK   256
#define NBA     1024           // blocks, pass A (min + histogram)
#define NBC     1024           // blocks, pass C (eval)
#define VPT     4              // float4 per thread per async stage (16KB LDS tile)

// ---------------- gfx1250 async global->LDS probe ----------------
#if defined(__has_builtin)
#if __has_builtin(__builtin_amdgcn_global_load_async_to_lds_b128) && \
    __has_builtin(__builtin_amdgcn_s_wait_asynccnt)
#define USE_ASYNC_LDS 1
#endif
#endif

// int4 vector type matching the async-builtin's expected pointee
typedef int v4i __attribute__((vector_size(16)));

// order-preserving float<->uint key (so integer atomicMin == float min, deterministic)
__device__ __forceinline__ unsigned f2key(float f) {
  unsigned u = __float_as_uint(f);
  return (u & 0x80000000u) ? ~u : (u | 0x80000000u);
}
__device__ __forceinline__ float key2f(unsigned k) {
  return (k & 0x80000000u) ? __uint_as_float(k ^ 0x80000000u) : __uint_as_float(~k);
}

// ---------------- kernel 0: reset workspace state ----------------
__global__ void k_init(unsigned* __restrict__ minbits, int* __restrict__ ihist) {
  int i = blockIdx.x * BLOCK + threadIdx.x;
  if (i < NI) ihist[i] = 0;
  if (i == NI) *minbits = 0xFFFFFFFFu;   // key of +inf
}

// ---------------- kernel A: fused global-min + offset histogram ----------------
__global__ void k_min_hist(const float* __restrict__ x, long long n,
                           unsigned* __restrict__ minbits, int* __restrict__ ihist) {
  __shared__ int   lh[NI];
  __shared__ float wmin[BLOCK / 32];
  for (int i = threadIdx.x; i < NI; i += BLOCK) lh[i] = 0;
  __syncthreads();

  float lmin = 3.402823466e38f;
  long long n4 = n >> 2;
  const float4* __restrict__ x4 = (const float4*)x;
  long long stride = (long long)gridDim.x * BLOCK;

  for (long long v = (long long)blockIdx.x * BLOCK + threadIdx.x; v < n4; v += stride) {
    if (v + 4 * stride < n4) __builtin_prefetch(&x4[v + 4 * stride], 0, 0);
    float4 p = x4[v];
    float a[4] = {p.x, p.y, p.z, p.w};
#pragma unroll
    for (int j = 0; j < 4; ++j) {
      float f = a[j];
      lmin = fminf(lmin, f);
      int b = (int)floorf(f * 5.0f) + IOFF;       // floor(x*5) + offset
      b = b < 0 ? 0 : (b > NI - 1 ? NI - 1 : b);
      atomicAdd(&lh[b], 1);                        // ds_add_u32
    }
  }
  // scalar tail (n not multiple of 4)
  for (long long v = (n4 << 2) + (long long)blockIdx.x * BLOCK + threadIdx.x; v < n; v += stride) {
    float f = x[v];
    lmin = fminf(lmin, f);
    int b = (int)floorf(f * 5.0f) + IOFF;
    b = b < 0 ? 0 : (b > NI - 1 ? NI - 1 : b);
    atomicAdd(&lh[b], 1);
  }

  // wave32 min tree, then cross-wave
  float m = lmin;
#pragma unroll
  for (int o = 16; o > 0; o >>= 1) m = fminf(m, __shfl_xor(m, o, 32));
  if ((threadIdx.x & 31) == 0) wmin[threadIdx.x >> 5] = m;
  __syncthreads();
  if (threadIdx.x == 0) {
    float bm = wmin[0];
#pragma unroll
    for (int i = 1; i < BLOCK / 32; ++i) bm = fminf(bm, wmin[i]);
    atomicMin(minbits, f2key(bm));                 // integer atomic -> deterministic
  }
  // flush LDS histogram (skip empty bins)
  for (int i = threadIdx.x; i < NI; i += BLOCK) {
    int c = lh[i];
    if (c) atomicAdd(&ihist[i], c);
  }
}

// ---------------- kernel B: fold 2048 -> 200 bins, build hist_add & g ----------------
__global__ void k_fold(const unsigned* __restrict__ minbits, const int* __restrict__ ihist,
                       long long n, float* __restrict__ vminn_out,
                       float* __restrict__ g, float* __restrict__ ha) {
  __shared__ float hist[NBINS];
  __shared__ float sha[HLEN];
  float fmin = key2f(*minbits);
  float vmin = floorf(fmin) - 1.0f;                // integer-valued
  int off = 5 * (int)vmin + IOFF;                  // final bin j <- intermediate bin j+off

  int j = threadIdx.x;
  if (j < NBINS) {
    int cnt = 0;
    if (j == 0) {                                  // low clamp collects k <= off
      int hi = off < NI - 1 ? off : NI - 1;
      for (int k = 0; k <= hi; ++k) cnt += ihist[k];
    } else if (j == NBINS - 1) {                   // high clamp collects k >= off+199
      int lo = off + NBINS - 1; if (lo < 0) lo = 0;
      for (int k = lo; k < NI; ++k) cnt += ihist[k];
    } else {
      int k = off + j;
      cnt = (k >= 0 && k < NI) ? ihist[k] : 0;
    }
    hist[j] = (float)cnt / (float)n;
  }
  __syncthreads();
  if (j < HLEN) {                                  // width-5 window sum
    float s = 0.0f;
#pragma unroll
    for (int t = 0; t < BPU; ++t) s += hist[j + t];
    sha[j] = s;
    ha[j]  = s;
  }
  __syncthreads();
  if (j < GLEN) g[j] = (sha[j + 1] - sha[j]) * 5.0f;
  if (j == 0) *vminn_out = vmin + 0.5f;
}

// ---------------- per-element loss ----------------
__device__ __forceinline__ float elem_loss(float f, float vminn,
                                           const float* lg, const float* lha) {
  float t = (f - vminn) * 5.0f;
  int i = (int)floorf(t);
  i = i < 0 ? 0 : (i > GLEN - 1 ? GLEN - 1 : i);
  float left = vminn + (float)i * 0.2f;
  float nl = (f - left) * lg[i] + lha[i] + 1e-8f;
  return __logf(nl);                               // v_log_f32 * ln2
}

// ---------------- kernel C: eval pass (async LDS staging) ----------------
__global__ void k_eval(const float* __restrict__ x, long long n,
                       const float* __restrict__ vminp, const float* __restrict__ gg,
                       const float* __restrict__ hha, float* __restrict__ parts) {
  __shared__ float lg[GLEN];
  __shared__ float lha[HLEN];
  __shared__ float wsum[BLOCK / 32];
#if USE_ASYNC_LDS
  __shared__ float4 buf[BLOCK * VPT];
#endif
  for (int i = threadIdx.x; i < GLEN; i += BLOCK) lg[i] = gg[i];
  for (int i = threadIdx.x; i < HLEN; i += BLOCK) lha[i] = hha[i];
  __syncthreads();

  float vminn = *vminp;
  float acc = 0.0f;
  long long n4 = n >> 2;
  const float4* __restrict__ x4 = (const float4*)x;

#if USE_ASYNC_LDS
  long long step = (long long)gridDim.x * BLOCK * VPT;
  for (long long base = (long long)blockIdx.x * BLOCK * VPT; base < n4; base += step) {
    // bulk async stage: VPT x b128 per thread into private LDS slots
#pragma unroll
    for (int j = 0; j < VPT; ++j) {
      long long v = base + (long long)j * BLOCK + threadIdx.x;
      if (v < n4)
        __builtin_amdgcn_global_load_async_to_lds_b128(
            (__attribute__((address_space(1))) v4i*)(x4 + v),
            (__attribute__((address_space(3))) v4i*)(buf + (j * BLOCK + threadIdx.x)),
            0, 0);
    }
    __builtin_amdgcn_s_wait_asynccnt(0);
#pragma unroll
    for (int j = 0; j < VPT; ++j) {
      long long v = base + (long long)j * BLOCK + threadIdx.x;
      if (v < n4) {
        float4 p = buf[j * BLOCK + threadIdx.x];   // own slot: no barrier needed
        acc += elem_loss(p.x, vminn, lg, lha);
        acc += elem_loss(p.y, vminn, lg, lha);
        acc += elem_loss(p.z, vminn, lg, lha);
        acc += elem_loss(p.w, vminn, lg, lha);
      }
    }
  }
#else
  long long stride = (long long)gridDim.x * BLOCK;
  for (long long v = (long long)blockIdx.x * BLOCK + threadIdx.x; v < n4; v += stride) {
    if (v + 4 * stride < n4) __builtin_prefetch(&x4[v + 4 * stride], 0, 0);
    float4 p = x4[v];
    acc += elem_loss(p.x, vminn, lg, lha);
    acc += elem_loss(p.y, vminn, lg, lha);
    acc += elem_loss(p.z, vminn, lg, lha);
    acc += elem_loss(p.w, vminn, lg, lha);
  }
#endif
  // scalar tail
  for (long long v = (n4 << 2) + (long long)blockIdx.x * BLOCK + threadIdx.x; v < n;
       v += (long long)gridDim.x * BLOCK)
    acc += elem_loss(x[v], vminn, lg, lha);

  // deterministic block reduction (wave32 tree + fixed-order cross-wave)
#pragma unroll
  for (int o = 16; o > 0; o >>= 1) acc += __shfl_xor(acc, o, 32);
  if ((threadIdx.x & 31) == 0) wsum[threadIdx.x >> 5] = acc;
  __syncthreads();
  if (threadIdx.x == 0) {
    float s = 0.0f;
#pragma unroll
    for (int i = 0; i < BLOCK / 32; ++i) s += wsum[i];
    parts[blockIdx.x] = s;
  }
}

// ---------------- kernel D: fixed-order final reduction ----------------
__global__ void k_final(const float* __restrict__ parts, int np, long long n,
                        float* __restrict__ out) {
  __shared__ double sd[BLOCK];
  double a = 0.0;
  for (int i = threadIdx.x; i < np; i += BLOCK) a += (double)parts[i];
  sd[threadIdx.x] = a;
  __syncthreads();
  for (int s = BLOCK / 2; s > 0; s >>= 1) {
    if (threadIdx.x < s) sd[threadIdx.x] += sd[threadIdx.x + s];
    __syncthreads();
  }
  if (threadIdx.x == 0) {
    float S = (float)sd[0];
    out[0] = S / (-0.69314718055994530942f) / (float)n;   // sum(log)/(-ln2)/n
  }
}

// ---------------- host launcher ----------------
extern "C" void kernel_launch(void* const* d_in, const int* in_sizes, int n_in,
                              void* d_out, int out_size, void* d_ws, size_t ws_size,
                              hipStream_t stream) {
  (void)n_in; (void)out_size; (void)ws_size;
  const float* x = (const float*)d_in[0];
  long long n = (long long)in_sizes[0];
  float* out = (float*)d_out;

  // d_ws layout (u32 units): [0] minkey | [16..16+2048) ihist |
  // [2064] vmin_new | [2080..+195) g | [2288..+196) hist_add | [2560..+NBC) partials
  unsigned* ws      = (unsigned*)d_ws;
  unsigned* minbits = ws;
  int*      ihist   = (int*)(ws + 16);
  float*    vminn   = (float*)(ws + 2064);
  float*    g       = (float*)(ws + 2080);
  float*    ha      = (float*)(ws + 2288);
  float*    parts   = (float*)(ws + 2560);

  k_init    <<<(NI / BLOCK) + 1, BLOCK, 0, stream>>>(minbits, ihist);
  k_min_hist<<<NBA, BLOCK, 0, stream>>>(x, n, minbits, ihist);
  k_fold    <<<1,   BLOCK, 0, stream>>>(minbits, ihist, n, vminn, g, ha);
  k_eval    <<<NBC, BLOCK, 0, stream>>>(x, n, vminn, g, ha, parts);
  k_final   <<<1,   BLOCK, 0, stream>>>(parts, NBC, n, out);
}